// Heatmap_AE_loss_30580167147966
// MI455X (gfx1250) — compile-verified
//
#include <hip/hip_runtime.h>
#include <hip/hip_bf16.h>

typedef float v2f __attribute__((ext_vector_type(2)));
typedef float v4f __attribute__((ext_vector_type(4)));
typedef float v8f __attribute__((ext_vector_type(8)));

#define Bn 48
#define Jn 17
#define Hn 128
#define Wn 128
#define Pn 30
#define HW (Hn * Wn)        // 16384
#define BJ (Bn * Jn)        // 816

// ---------------------------------------------------------------------------
// Kernel 1: streaming pass. One block per (b,j) heatmap tile (16384 floats of
// pred + 16384 of gt). 8 waves/block, each wave covers 2048 elements in 16
// iterations of 128 elements (float4 loads -> global_load_b128, 512B/instr).
// Per iteration, four V_WMMA_F32_16X16X4_F32 ops (A = ones) fold the wave's
// values into two independent f32 accumulator chains per quantity:
//   acc_sq* += sum of (pred-gt)^2 chunk, acc_gt* += sum of gt chunk.
// Writes {sum_sq, sum_gt} per (b,j) to workspace.
// ---------------------------------------------------------------------------
__global__ __launch_bounds__(256) void k_mse_partial(
    const float* __restrict__ outputs,
    const float* __restrict__ heatmaps,
    float* __restrict__ ws)
{
    const int bid  = blockIdx.x;            // 0..815
    const int b    = bid / Jn;
    const int j    = bid % Jn;
    const float* __restrict__ pred = outputs  + ((size_t)b * (2 * Jn) + j) * HW;
    const float* __restrict__ gt   = heatmaps + ((size_t)b * Jn       + j) * HW;

    const int lane = threadIdx.x & 31;
    const int wave = threadIdx.x >> 5;
    const int base = wave * 2048;

    v8f acc_sq0 = {}, acc_sq1 = {};
    v8f acc_gt0 = {}, acc_gt1 = {};
    v2f ones;
    ones.x = 1.0f; ones.y = 1.0f;

    for (int i = 0; i < 16; ++i) {
        const int e = base + i * 128 + lane * 4;
        v4f p = *(const v4f*)(pred + e);
        v4f g = *(const v4f*)(gt   + e);
        // Run-ahead prefetch (8 KB beyond the in-flight load window); TH is
        // speculative so an over-run at the tile end is silently dropped.
        __builtin_prefetch(pred + e + 2048, 0, 0);   // global_prefetch_b8
        __builtin_prefetch(gt   + e + 2048, 0, 0);

        v4f d  = p - g;
        v4f sq = d * d;
        v2f sq_lo; sq_lo.x = sq.x; sq_lo.y = sq.y;
        v2f sq_hi; sq_hi.x = sq.z; sq_hi.y = sq.w;
        v2f g_lo;  g_lo.x  = g.x;  g_lo.y  = g.y;
        v2f g_hi;  g_hi.x  = g.z;  g_hi.y  = g.w;

        // D[m,n] = sum_k B[k,n] + C[m,n]  (A = ones) -> wave-wide f32 sum fold.
        // Two independent accumulator chains per quantity for WMMA ILP.
        acc_sq0 = __builtin_amdgcn_wmma_f32_16x16x4_f32(
            false, ones, false, sq_lo, (short)0, acc_sq0, false, false);
        acc_gt0 = __builtin_amdgcn_wmma_f32_16x16x4_f32(
            false, ones, false, g_lo,  (short)0, acc_gt0, false, false);
        acc_sq1 = __builtin_amdgcn_wmma_f32_16x16x4_f32(
            false, ones, false, sq_hi, (short)0, acc_sq1, false, false);
        acc_gt1 = __builtin_amdgcn_wmma_f32_16x16x4_f32(
            false, ones, false, g_hi,  (short)0, acc_gt1, false, false);
    }

    // D VGPR0: every lane holds a column total; each column appears twice
    // (rows M=0 and M=8 are identical) -> full-wave shuffle sum, then halve.
    float vsq = acc_sq0[0] + acc_sq1[0];
    float vgt = acc_gt0[0] + acc_gt1[0];
    for (int off = 16; off >= 1; off >>= 1) {
        vsq += __shfl_xor(vsq, off, 32);
        vgt += __shfl_xor(vgt, off, 32);
    }

    __shared__ float s_sq[8];
    __shared__ float s_gt[8];
    if (lane == 0) { s_sq[wave] = vsq * 0.5f; s_gt[wave] = vgt * 0.5f; }
    __syncthreads();
    if (threadIdx.x == 0) {
        float tsq = 0.0f, tgt = 0.0f;
        for (int w = 0; w < 8; ++w) { tsq += s_sq[w]; tgt += s_gt[w]; }
        ws[bid * 2 + 0] = tsq;
        ws[bid * 2 + 1] = tgt;
    }
}

// ---------------------------------------------------------------------------
// Kernel 2: finalize heatmap MSE over the 816 (b,j) partials.
//   mask = (sum_gt > 0); hm = sum(mask * sum_sq/16384) / sum(mask)
// ---------------------------------------------------------------------------
__global__ __launch_bounds__(256) void k_hm_final(
    const float* __restrict__ ws, float* __restrict__ out)
{
    __shared__ float s_num[256];
    __shared__ float s_den[256];
    float num = 0.0f, den = 0.0f;
    for (int i = threadIdx.x; i < BJ; i += 256) {
        const float m = (ws[i * 2 + 1] > 0.0f) ? 1.0f : 0.0f;
        num += ws[i * 2 + 0] * (1.0f / (float)HW) * m;
        den += m;
    }
    s_num[threadIdx.x] = num;
    s_den[threadIdx.x] = den;
    __syncthreads();
    for (int s = 128; s >= 1; s >>= 1) {
        if (threadIdx.x < s) {
            s_num[threadIdx.x] += s_num[threadIdx.x + s];
            s_den[threadIdx.x] += s_den[threadIdx.x + s];
        }
        __syncthreads();
    }
    if (threadIdx.x == 0)
        out[0] = (s_den[0] > 0.0f) ? (s_num[0] / s_den[0]) : 0.0f;
}

// ---------------------------------------------------------------------------
// Kernel 3: associative-embedding push/pull, one block per batch.
// Thread p (< 30) gathers its person's 17 tags, computes mean / pull; thread 0
// then does the 30x30 push pass. Tiny vs the streaming kernel.
// ---------------------------------------------------------------------------
__global__ __launch_bounds__(32) void k_ae(
    const float* __restrict__ outputs,
    const long long* __restrict__ joints,   // int64 (B,P,J,2)
    float* __restrict__ out)
{
    const int b = blockIdx.x;
    const float* __restrict__ tags =
        outputs + ((size_t)b * (2 * Jn) + Jn) * HW;   // tags[b] flat, length J*H*W

    __shared__ float s_m[Pn];
    __shared__ float s_pull[Pn];
    __shared__ int   s_valid[Pn];

    const int p = threadIdx.x;
    if (p < Pn) {
        float tvals[Jn];
        float vis[Jn];
        float sum_tv = 0.0f, cnt = 0.0f;
        for (int j = 0; j < Jn; ++j) {
            const size_t ji = (((size_t)b * Pn + p) * Jn + j) * 2;
            const long long idx = joints[ji + 0];
            const long long v   = joints[ji + 1];
            const float t  = tags[idx];
            const float vf = (v > 0) ? 1.0f : 0.0f;
            tvals[j] = t;
            vis[j]   = vf;
            sum_tv  += t * vf;
            cnt     += vf;
        }
        const bool  valid = (cnt > 0.0f);
        const float safe  = valid ? cnt : 1.0f;
        const float m     = sum_tv / safe;
        float pull = 0.0f;
        for (int j = 0; j < Jn; ++j) {
            const float d = tvals[j] - m;
            pull += vis[j] * d * d;
        }
        pull /= safe;
        s_m[p]     = m;
        s_pull[p]  = valid ? pull : 0.0f;
        s_valid[p] = valid ? 1 : 0;
    }
    __syncthreads();

    if (threadIdx.x == 0) {
        float num_tags = 0.0f, pull_sum = 0.0f;
        for (int q = 0; q < Pn; ++q) {
            num_tags += (float)s_valid[q];
            pull_sum += s_pull[q];
        }
        float push_raw = 0.0f;
        for (int a = 0; a < Pn; ++a) {
            if (!s_valid[a]) continue;
            for (int c = 0; c < Pn; ++c) {
                if (!s_valid[c]) continue;
                const float d = s_m[a] - s_m[c];
                push_raw += expf(-d * d);
            }
        }
        push_raw -= num_tags;
        const float denom = (num_tags - 1.0f) * num_tags;
        const float push  = (num_tags >= 2.0f)
            ? push_raw / ((denom > 0.0f) ? denom : 1.0f) * 0.5f
            : 0.0f;
        const float pull  = pull_sum / fmaxf(num_tags, 1.0f);
        out[1 + b]        = push;   // push[48] at out[1..48]
        out[1 + Bn + b]   = pull;   // pull[48] at out[49..96]
    }
}

// ---------------------------------------------------------------------------
extern "C" void kernel_launch(void* const* d_in, const int* in_sizes, int n_in,
                              void* d_out, int out_size, void* d_ws, size_t ws_size,
                              hipStream_t stream)
{
    const float*     outputs  = (const float*)d_in[0];
    const float*     heatmaps = (const float*)d_in[1];
    const long long* joints   = (const long long*)d_in[2];  // int64 per reference
    float* out = (float*)d_out;
    float* ws  = (float*)d_ws;   // 816*2 floats

    k_mse_partial<<<BJ, 256, 0, stream>>>(outputs, heatmaps, ws);
    k_hm_final<<<1, 256, 0, stream>>>(ws, out);
    k_ae<<<Bn, 32, 0, stream>>>(outputs, joints, out);
}